// GaussianRenderer_34488587386978
// MI455X (gfx1250) — compile-verified
//
#include <hip/hip_runtime.h>
#include <math.h>

// Gaussian splat renderer for MI455X (gfx1250, wave32).
// Quadratic-form exponents evaluated on the matrix pipe via chained
// V_WMMA_F32_16X16X4_F32 (16 gaussians x 16 pixels per pair of WMMAs).
// log2(e) is folded into the coefficients so WMMA output feeds the bare
// hardware v_exp_f32 (2^x) with no intervening VALU. Colors broadcast via
// ds_bpermute; halves merged with one SWAPX16 exchange per 16-gaussian batch.

#define NG   512
#define IMG_H 256
#define IMG_W 256
#define NBATCH (NG / 16)

typedef __attribute__((ext_vector_type(2))) float v2f;
typedef __attribute__((ext_vector_type(8))) float v8f;

// ---------------------------------------------------------------------------
// Kernel 1: per-gaussian projection + coefficient build + stable depth sort.
// One block of 512 threads; O(N^2) rank sort in LDS (stable, matches argsort).
// Coefficients of (expo + ln(opacity*norm)) * log2(e) as a 6-term polynomial:
//   e2 = c0*x^2 + c1*x*y + c2*y^2 + c3*x + c4*y + c5,  alpha = 2^e2
// Invalid gaussians get c5 = -1e30 -> alpha = 0.
// ---------------------------------------------------------------------------
__global__ __launch_bounds__(NG) void gs_preprocess(
    const float* __restrict__ means3D,   // [512,3]
    const float* __restrict__ covs3d,    // [512,3,3]
    const float* __restrict__ colors,    // [512,3]
    const float* __restrict__ opac,      // [512]
    const float* __restrict__ Km,        // [3,3]
    const float* __restrict__ Rm,        // [3,3]
    const float* __restrict__ tv,        // [3]
    float4* __restrict__ set0,           // [512] (c0,c1,c2,c3) sorted
    float4* __restrict__ set1,           // [512] (c4,c5,0,0)   sorted
    float4* __restrict__ colw)           // [512] (r,g,b,0)     sorted
{
    __shared__ float sdepth[NG];
    const int n = threadIdx.x;

    // --- load gaussian ---
    const float m0 = means3D[n * 3 + 0];
    const float m1 = means3D[n * 3 + 1];
    const float m2 = means3D[n * 3 + 2];

    float R[3][3], S[3][3];
#pragma unroll
    for (int i = 0; i < 3; ++i)
#pragma unroll
        for (int j = 0; j < 3; ++j) {
            R[i][j] = Rm[i * 3 + j];
            S[i][j] = covs3d[n * 9 + i * 3 + j];
        }

    // cam = R*m + t
    const float cam0 = R[0][0] * m0 + R[0][1] * m1 + R[0][2] * m2 + tv[0];
    const float cam1 = R[1][0] * m0 + R[1][1] * m1 + R[1][2] * m2 + tv[1];
    const float cam2 = R[2][0] * m0 + R[2][1] * m1 + R[2][2] * m2 + tv[2];

    const float Z     = cam2;
    const float depth = fmaxf(Z, 1.0f);

    // screen = K*cam ; means2D = screen.xy / screen.z
    const float s0 = Km[0] * cam0 + Km[1] * cam1 + Km[2] * cam2;
    const float s1 = Km[3] * cam0 + Km[4] * cam1 + Km[5] * cam2;
    const float s2 = Km[6] * cam0 + Km[7] * cam1 + Km[8] * cam2;
    const float mx = s0 / s2;
    const float my = s1 / s2;

    const float fx = Km[0], fy = Km[4];
    const float invZ  = 1.0f / Z;
    const float J00 = fx * invZ;
    const float J02 = -fx * cam0 * invZ * invZ;
    const float J11 = fy * invZ;
    const float J12 = -fy * cam1 * invZ * invZ;

    // M = R * S * R^T
    float TMP[3][3], M[3][3];
#pragma unroll
    for (int i = 0; i < 3; ++i)
#pragma unroll
        for (int k = 0; k < 3; ++k)
            TMP[i][k] = R[i][0] * S[0][k] + R[i][1] * S[1][k] + R[i][2] * S[2][k];
#pragma unroll
    for (int i = 0; i < 3; ++i)
#pragma unroll
        for (int l = 0; l < 3; ++l)
            M[i][l] = TMP[i][0] * R[l][0] + TMP[i][1] * R[l][1] + TMP[i][2] * R[l][2];

    // V = J * M * J^T  (J = [[J00,0,J02],[0,J11,J12]])
    float u0[3], u1[3];
#pragma unroll
    for (int k = 0; k < 3; ++k) {
        u0[k] = J00 * M[0][k] + J02 * M[2][k];
        u1[k] = J11 * M[1][k] + J12 * M[2][k];
    }
    const float a = u0[0] * J00 + u0[2] * J02 + 1e-4f;
    const float b = u0[1] * J11 + u0[2] * J12;
    const float c = u1[1] * J11 + u1[2] * J12 + 1e-4f;

    const float det  = a * c - b * b;
    const float idet = 1.0f / det;
    const float ia =  c * idet;
    const float ib = -b * idet;
    const float ic =  a * idet;
    const float norm = 0.15915494309189535f / sqrtf(det);  // 1/(2*pi*sqrt(det))

    const bool  valid = (depth > 1.0f) && (depth < 50.0f);
    const float sc    = opac[n] * norm;

    float c0, c1, c2, c3, c4, c5;
    if (valid) {
        const float ls2 = (sc > 0.0f) ? __log2f(sc) : -1e30f;  // log2 scale
        const float k2  = 1.4426950408889634f;                 // log2(e)
        c0 = (-0.5f * ia) * k2;
        c1 = (-ib) * k2;
        c2 = (-0.5f * ic) * k2;
        c3 = (ia * mx + ib * my) * k2;
        c4 = (ib * mx + ic * my) * k2;
        c5 = (-0.5f * (ia * mx * mx + 2.0f * ib * mx * my + ic * my * my)) * k2 + ls2;
    } else {
        c0 = c1 = c2 = c3 = c4 = 0.0f;
        c5 = -1e30f;
    }

    // --- stable ascending rank sort by depth (matches jnp.argsort) ---
    sdepth[n] = depth;
    __syncthreads();
    int rank = 0;
    const float dn = depth;
    for (int m = 0; m < NG; ++m) {
        const float dm = sdepth[m];
        rank += (dm < dn) || (dm == dn && m < n);
    }

    set0[rank] = make_float4(c0, c1, c2, c3);
    set1[rank] = make_float4(c4, c5, 0.0f, 0.0f);
    colw[rank] = make_float4(colors[n * 3 + 0], colors[n * 3 + 1],
                             colors[n * 3 + 2], 0.0f);
}

// ---------------------------------------------------------------------------
// Kernel 2: render. One wave per 16-pixel row tile (4096 tiles).
// Per batch of 16 gaussians:
//   D = A(16g x 4feat) x B(4feat x 16pix) chained over K=8 features
//   via two V_WMMA_F32_16X16X4_F32 -> e2[g][pix], alpha = v_exp_f32(e2).
// D layout: lane L (pix = L&15) holds gaussians r+8*(L>=16) in v[r], r=0..7.
// Each lane composites its 8 gaussians serially, halves merged by shfl_xor 16.
// Reference uses inclusive cumprod: w = alpha*(1-alpha)*T_exclusive.
// ---------------------------------------------------------------------------
__global__ __launch_bounds__(256) void gs_render(
    const float4* __restrict__ set0,
    const float4* __restrict__ set1,
    const float4* __restrict__ colw,
    float* __restrict__ out)             // [256,256,3]
{
    const int lane = threadIdx.x & 31;
    const int wave = threadIdx.x >> 5;
    const int tile = blockIdx.x * 8 + wave;       // 0..4095
    const int y    = tile >> 4;                   // row
    const int x0   = (tile & 15) << 4;            // tile start column
    const int nn   = lane & 15;                   // pixel within tile / gaussian slot
    const int hi   = lane >> 4;                   // wave half
    const bool hib = (hi != 0);

    const float xf = (float)(x0 + nn);
    const float yf = (float)y;

    // B operand (constant over the gaussian loop).
    // features: K0=x^2 K1=x*y K2=y^2 K3=x | K4=y K5=1 K6=0 K7=0
    v2f b1, b2;
    b1.x = hib ? yf * yf : xf * xf;
    b1.y = hib ? xf      : xf * yf;
    b2.x = hib ? 0.0f    : yf;
    b2.y = hib ? 0.0f    : 1.0f;

    float T_run = 1.0f, accR = 0.0f, accG = 0.0f, accB = 0.0f;

    for (int batch = 0; batch < NBATCH; ++batch) {
        const int g = batch * 16 + nn;
        const float4 s0 = set0[g];   // (c0,c1,c2,c3) of gaussian g
        const float4 s1 = set1[g];   // (c4,c5,0,0)
        const float4 cl = colw[g];   // (r,g,b,0)

        // A operand: lane holds K{0,1} (half 0) or K{2,3} (half 1) of row M=nn
        v2f a1, a2;
        a1.x = hib ? s0.z : s0.x;
        a1.y = hib ? s0.w : s0.y;
        a2.x = hib ? s1.z : s1.x;
        a2.y = hib ? s1.w : s1.y;

        v8f cz = {0.f, 0.f, 0.f, 0.f, 0.f, 0.f, 0.f, 0.f};
        v8f d = __builtin_amdgcn_wmma_f32_16x16x4_f32(
            false, a1, false, b1, (short)0, cz, false, false);
        d = __builtin_amdgcn_wmma_f32_16x16x4_f32(
            false, a2, false, b2, (short)0, d, false, false);

        // serial composite of this half's 8 gaussians for this pixel
        float t_p = 1.0f, cr = 0.0f, cg = 0.0f, cb = 0.0f;
        const int sbase = hi << 3;
#pragma unroll
        for (int r = 0; r < 8; ++r) {
            const int   src  = sbase + r;              // lane holding color of M=r(+8)
            const float colr = __shfl(cl.x, src, 32);
            const float colg = __shfl(cl.y, src, 32);
            const float colb = __shfl(cl.z, src, 32);
            const float alpha = __builtin_amdgcn_exp2f(d[r]);  // bare v_exp_f32
            const float one   = 1.0f - alpha;
            const float w     = alpha * one * t_p;     // inclusive cumprod semantics
            cr += w * colr;
            cg += w * colg;
            cb += w * colb;
            t_p *= one;
        }

        // merge halves: low half = gaussians 0..7, high half = 8..15
        const float ot  = __shfl_xor(t_p, 16, 32);
        const float ocr = __shfl_xor(cr, 16, 32);
        const float ocg = __shfl_xor(cg, 16, 32);
        const float ocb = __shfl_xor(cb, 16, 32);

        const float bR = hib ? (ocr + ot * cr) : (cr + t_p * ocr);
        const float bG = hib ? (ocg + ot * cg) : (cg + t_p * ocg);
        const float bB = hib ? (ocb + ot * cb) : (cb + t_p * ocb);
        const float tb = t_p * ot;

        accR += T_run * bR;
        accG += T_run * bG;
        accB += T_run * bB;
        T_run *= tb;
    }

    if (lane < 16) {
        const int o = (y * IMG_W + (x0 + nn)) * 3;
        out[o + 0] = accR;
        out[o + 1] = accG;
        out[o + 2] = accB;
    }
}

// ---------------------------------------------------------------------------
extern "C" void kernel_launch(void* const* d_in, const int* in_sizes, int n_in,
                              void* d_out, int out_size, void* d_ws, size_t ws_size,
                              hipStream_t stream) {
    const float* means3D = (const float*)d_in[0];
    const float* covs3d  = (const float*)d_in[1];
    const float* colors  = (const float*)d_in[2];
    const float* opac    = (const float*)d_in[3];
    const float* Km      = (const float*)d_in[4];
    const float* Rm      = (const float*)d_in[5];
    const float* tv      = (const float*)d_in[6];

    float4* set0 = (float4*)d_ws;        // 512 * 16B
    float4* set1 = set0 + NG;            // 512 * 16B
    float4* colw = set1 + NG;            // 512 * 16B  (24 KB total, lives in L2)

    gs_preprocess<<<1, NG, 0, stream>>>(means3D, covs3d, colors, opac,
                                        Km, Rm, tv, set0, set1, colw);

    // 4096 tiles (16 pixels each), 8 waves (256 threads) per block
    gs_render<<<512, 256, 0, stream>>>(set0, set1, colw, (float*)d_out);
}